// GIN_29257317220564
// MI455X (gfx1250) — compile-verified
//
#include <hip/hip_runtime.h>

typedef __attribute__((ext_vector_type(2))) float v2f;
typedef __attribute__((ext_vector_type(8))) float v8f;

#define N_NODES   50000
#define N_EDGES   800000
#define KDIM      128          // input feature dim for every layer
#define LDS_STRIDE 132         // 128 + 4 pad -> bank-conflict-free column reads

// Guaranteed hardware float atomic (GLOBAL_ATOMIC_ADD_F32, non-returning):
// avoids any CAS-loop lowering of plain atomicAdd(float*).
__device__ __forceinline__ void fatomic_add(float* p, float v) {
    unsafeAtomicAdd(p, v);
}

// ---------------------------------------------------------------- zero buffer
__global__ void gin_zero_kernel(float4* __restrict__ p, int n4) {
    int i = blockIdx.x * blockDim.x + threadIdx.x;
    if (i < n4) p[i] = make_float4(0.f, 0.f, 0.f, 0.f);
}

// ------------------------------------------------------- edge scatter-add agg
// One wave (32 lanes) per edge; each lane handles 4 consecutive floats
// (32 * 4 = 128 features). Gathers hit L2 (node table is 25.6 MB << 192 MB L2).
__global__ void gin_scatter_add_kernel(const float* __restrict__ x,
                                       const int*   __restrict__ src,
                                       const int*   __restrict__ dst,
                                       float*       __restrict__ agg) {
    int gid  = blockIdx.x * blockDim.x + threadIdx.x;
    int e    = gid >> 5;
    if (e >= N_EDGES) return;
    int lane = gid & 31;

    int s = src[e];
    int d = dst[e];

    const float4 v = *(const float4*)(x + (size_t)s * KDIM + lane * 4);
    float* o = agg + (size_t)d * KDIM + lane * 4;
    fatomic_add(o + 0, v.x);
    fatomic_add(o + 1, v.y);
    fatomic_add(o + 2, v.z);
    fatomic_add(o + 3, v.w);
}

// ------------------------------------------- fused (xa+xb) @ W^T + b, ReLU
// Grid: one block per 16-node tile (50000/16 = 3125 exact).
// Block: 32 * (outF/16) threads; wave w computes output columns [16w, 16w+16).
// K loop: 32 x V_WMMA_F32_16X16X4_F32 (fp32-exact).
__global__ void gin_gemm_relu_kernel(const float* __restrict__ xa,
                                     const float* __restrict__ xb,
                                     const float* __restrict__ W,     // [outF][KDIM]
                                     const float* __restrict__ bias,  // [outF]
                                     float*       __restrict__ out,   // [N][outF]
                                     int outF) {
    __shared__ float As[16 * LDS_STRIDE];

    const int nodeBase = blockIdx.x * 16;
    const int tid      = threadIdx.x;
    const int nThreads = blockDim.x;

    // Stage A tile = (xa + xb)[nodeBase:nodeBase+16, 0:128] into LDS.
    for (int idx = tid; idx < 16 * KDIM; idx += nThreads) {
        int r = idx >> 7;        // / 128
        int k = idx & (KDIM - 1);
        size_t g = (size_t)(nodeBase + r) * KDIM + k;
        As[r * LDS_STRIDE + k] = xa[g] + xb[g];
    }
    __syncthreads();

    const int wave  = tid >> 5;
    const int lane  = tid & 31;
    const int jBase = wave * 16;
    const int nl    = lane & 15;                 // M (for A) / N (for B) index
    const int koff  = (lane < 16) ? 0 : 2;       // ISA 16x4 f32 A/B lane layout

    const float* arow = As + nl * LDS_STRIDE;            // A row m = nl
    const float* wrow = W + (size_t)(jBase + nl) * KDIM; // B[k][n] = W[jBase+n][k]

    v8f c = {};
    #pragma unroll
    for (int k0 = 0; k0 < KDIM; k0 += 4) {
        v2f a, b;
        a.x = arow[k0 + koff];
        a.y = arow[k0 + koff + 1];
        b.x = wrow[k0 + koff];
        b.y = wrow[k0 + koff + 1];
        // (neg_a, A, neg_b, B, c_mod, C, reuse_a, reuse_b)
        c = __builtin_amdgcn_wmma_f32_16x16x4_f32(false, a, false, b,
                                                  (short)0, c, false, false);
    }

    // C/D layout: lanes 0-15 -> N=lane, M=v; lanes 16-31 -> N=lane-16, M=v+8.
    const int j     = jBase + nl;
    const float bj  = bias[j];
    const int mBase = (lane < 16) ? 0 : 8;
    #pragma unroll
    for (int v = 0; v < 8; ++v) {
        float r = c[v] + bj;
        out[(size_t)(nodeBase + mBase + v) * outF + j] = fmaxf(r, 0.f);
    }
}

// ---------------------------------------------------------------- launcher
extern "C" void kernel_launch(void* const* d_in, const int* in_sizes, int n_in,
                              void* d_out, int out_size, void* d_ws, size_t ws_size,
                              hipStream_t stream) {
    const float* in_feat = (const float*)d_in[0];
    const float* W1      = (const float*)d_in[1];
    const float* b1      = (const float*)d_in[2];
    const float* W2      = (const float*)d_in[3];
    const float* b2      = (const float*)d_in[4];
    const float* W3      = (const float*)d_in[5];
    const float* b3      = (const float*)d_in[6];
    const int*   src     = (const int*)d_in[7];
    const int*   dst     = (const int*)d_in[8];
    float*       out     = (float*)d_out;

    char* ws = (char*)d_ws;
    const size_t bufBytes = (size_t)N_NODES * KDIM * sizeof(float); // 25.6 MB
    float* agg = (float*)(ws);
    float* h1  = (float*)(ws + bufBytes);
    float* h2  = (float*)(ws + 2 * bufBytes);

    const int n4 = N_NODES * KDIM / 4;
    dim3 zgrid((n4 + 255) / 256);
    dim3 sgrid((N_EDGES * 32 + 255) / 256);
    dim3 ggrid(N_NODES / 16);   // 3125

    // ---- layer 1: in_feat -> h1
    gin_zero_kernel<<<zgrid, 256, 0, stream>>>((float4*)agg, n4);
    gin_scatter_add_kernel<<<sgrid, 256, 0, stream>>>(in_feat, src, dst, agg);
    gin_gemm_relu_kernel<<<ggrid, 256, 0, stream>>>(in_feat, agg, W1, b1, h1, 128);

    // ---- layer 2: h1 -> h2
    gin_zero_kernel<<<zgrid, 256, 0, stream>>>((float4*)agg, n4);
    gin_scatter_add_kernel<<<sgrid, 256, 0, stream>>>(h1, src, dst, agg);
    gin_gemm_relu_kernel<<<ggrid, 256, 0, stream>>>(h1, agg, W2, b2, h2, 128);

    // ---- layer 3: h2 -> out (64 classes -> 4 waves per block)
    gin_zero_kernel<<<zgrid, 256, 0, stream>>>((float4*)agg, n4);
    gin_scatter_add_kernel<<<sgrid, 256, 0, stream>>>(h2, src, dst, agg);
    gin_gemm_relu_kernel<<<ggrid, 128, 0, stream>>>(h2, agg, W3, b3, out, 64);
}